// Sinkhorn_22703197127350
// MI455X (gfx1250) — compile-verified
//
#include <hip/hip_runtime.h>
#include <hip/hip_bf16.h>
#include <math.h>

// Sinkhorn OT for MI455X (gfx1250, wave32).
// Strategy: never materialize the 256MB cost matrix during iterations (it
// exceeds the 192MB L2); recompute C_ij = |x|^2 + |y|^2 - 2<x,y> on the fly
// with V_WMMA_F32_16X16X4_F32 (fp32, K=4 fits D=3 padded) and do an online
// logsumexp per row strip. Only the final pi/C materialization touches HBM
// at scale (512MB stores ~= 22us at 23.3 TB/s).

#define N_BATCH 4
#define P1_PTS 4096
#define P2_PTS 4096
#define EPS_F 0.1f
#define INV_EPS_F 10.0f
#define MAX_ITER 50
#define THRESH_F 0.1f
#define WAVES_PER_BLOCK 4

typedef __attribute__((ext_vector_type(2))) float v2f;
typedef __attribute__((ext_vector_type(8))) float v8f;

__device__ __forceinline__ v8f wmma_dot16x16(v2f a, v2f b) {
    v8f c = {0.f, 0.f, 0.f, 0.f, 0.f, 0.f, 0.f, 0.f};
    // D = A(16x4, f32) * B(4x16, f32) + 0  -> 16x16 f32 dot-product tile
    return __builtin_amdgcn_wmma_f32_16x16x4_f32(
        /*neg_a=*/false, a, /*neg_b=*/false, b,
        /*c_mod=*/(short)0, c, /*reuse_a=*/false, /*reuse_b=*/false);
}

// ---------------------------------------------------------------------------
// init: zero u, v, err slots, done flag, and the output cost accumulator.
// ---------------------------------------------------------------------------
__global__ void sinkhorn_init(float* __restrict__ u, float* __restrict__ v,
                              float* __restrict__ err, int* __restrict__ done,
                              float* __restrict__ cost) {
    const int t = blockIdx.x * blockDim.x + threadIdx.x;
    if (t < N_BATCH * P1_PTS) u[t] = 0.0f;
    if (t < N_BATCH * P2_PTS) v[t] = 0.0f;
    if (t < MAX_ITER) err[t] = 0.0f;
    if (t == 0) { done[0] = 0; cost[0] = 0.0f; }
}

// ---------------------------------------------------------------------------
// One half-iteration: logsumexp over the "col" axis of M for a 16-row strip,
// then dual update  u_new = eps*(log(w + 1e-8) - lse) + u   (in place).
// Row pass: (a=x, b=y, dual=u, other=v). Col pass: roles swapped (C symmetric
// under swap), no err accumulation.
// ---------------------------------------------------------------------------
__global__ void __launch_bounds__(WAVES_PER_BLOCK * 32)
sinkhorn_lse_pass(const float* __restrict__ a_pts,   // [N, PA, 3]
                  const float* __restrict__ b_pts,   // [N, PB, 3]
                  const float* __restrict__ w_row,   // [N, PA] (mu or nu)
                  const float* __restrict__ v_col,   // [N, PB] other dual
                  float* __restrict__ u_row,         // [N, PA] dual (in/out)
                  float* __restrict__ err_slot,      // &err[iter]
                  const int* __restrict__ done_flag,
                  int PA, int PB, int accumulate_err) {
    if (*done_flag) return;  // wave-uniform: converged -> freeze duals

    const int n    = blockIdx.y;
    const int wave = threadIdx.x >> 5;
    const int lane = threadIdx.x & 31;
    const int half = lane >> 4;   // 0: K=0/1 carrier, 1: K=2/3 carrier
    const int l15  = lane & 15;
    const int i0   = (blockIdx.x * WAVES_PER_BLOCK + wave) * 16;

    // A fragment (16x4 fp32 layout): lanes 0-15 hold K0/K1, lanes 16-31 K2/K3.
    const float* ap = a_pts + ((size_t)(n * PA + i0 + l15)) * 3;
    v2f afrag;
    afrag.x = half ? ap[2] : ap[0];
    afrag.y = half ? 0.0f  : ap[1];

    // Per-accumulator row state (C/D layout: VGPR g -> row i0 + g + half*8).
    float u8[8], xn8[8], m8[8], s8[8];
#pragma unroll
    for (int g = 0; g < 8; ++g) {
        const int row  = i0 + g + half * 8;
        const float* p = a_pts + ((size_t)(n * PA + row)) * 3;
        xn8[g] = p[0] * p[0] + p[1] * p[1] + p[2] * p[2];
        u8[g]  = u_row[n * PA + row];
        m8[g]  = -3.0e38f;
        s8[g]  = 0.0f;
    }

    for (int j0 = 0; j0 < PB; j0 += 16) {
        const int col   = j0 + l15;
        const float* bp = b_pts + ((size_t)(n * PB + col)) * 3;
        const float b0 = bp[0], b1 = bp[1], b2 = bp[2];
        v2f bfrag;
        bfrag.x = half ? b2   : b0;
        bfrag.y = half ? 0.0f : b1;
        const float yn = b0 * b0 + b1 * b1 + b2 * b2;
        const float vc = v_col[n * PB + col];

        const v8f dotc = wmma_dot16x16(afrag, bfrag);

#pragma unroll
        for (int g = 0; g < 8; ++g) {
            const float Cij = xn8[g] + yn - 2.0f * dotc[g];
            const float Mv  = (u8[g] + vc - Cij) * INV_EPS_F;
            const float mn  = fmaxf(m8[g], Mv);
            s8[g] = s8[g] * __expf(m8[g] - mn) + __expf(Mv - mn);
            m8[g] = mn;
        }
    }

    // Merge the online LSE across the 16 lanes of each half-wave
    // (xor masks < 16 never cross the half boundary).
#pragma unroll
    for (int off = 1; off < 16; off <<= 1) {
#pragma unroll
        for (int g = 0; g < 8; ++g) {
            const float mo = __shfl_xor(m8[g], off, 32);
            const float so = __shfl_xor(s8[g], off, 32);
            const float mn = fmaxf(m8[g], mo);
            s8[g] = s8[g] * __expf(m8[g] - mn) + so * __expf(mo - mn);
            m8[g] = mn;
        }
    }

    if (l15 == 0) {  // lane 0 owns rows i0..i0+7, lane 16 owns i0+8..i0+15
        float errloc = 0.0f;
#pragma unroll
        for (int g = 0; g < 8; ++g) {
            const int row   = i0 + g + half * 8;
            const float lse = m8[g] + __logf(s8[g]);
            const float un  =
                EPS_F * (__logf(w_row[n * PA + row] + 1e-8f) - lse) + u8[g];
            u_row[n * PA + row] = un;
            errloc += fabsf(un - u8[g]);
        }
        if (accumulate_err) atomicAdd(err_slot, errloc);
    }
}

// ---------------------------------------------------------------------------
// Convergence flag: runs AFTER both dual updates of iteration `it`, matching
// the reference's "apply update, then mark done" semantics.
// ---------------------------------------------------------------------------
__global__ void sinkhorn_flag(const float* __restrict__ err, int it,
                              int* __restrict__ done, float thresh_total) {
    if (threadIdx.x == 0 && blockIdx.x == 0) {
        if (!done[0] && err[it] < thresh_total) done[0] = 1;
    }
}

// ---------------------------------------------------------------------------
// Final: materialize pi = exp(M(u,v)) and C, accumulate cost = sum(pi * C).
// ---------------------------------------------------------------------------
__global__ void __launch_bounds__(32)
sinkhorn_final(const float* __restrict__ x_pts, const float* __restrict__ y_pts,
               const float* __restrict__ u, const float* __restrict__ v,
               float* __restrict__ pi_out, float* __restrict__ c_out,
               float* __restrict__ cost) {
    const int n    = blockIdx.z;
    const int i0   = blockIdx.y * 16;
    const int j0   = blockIdx.x * 16;
    const int lane = threadIdx.x;
    const int half = lane >> 4;
    const int l15  = lane & 15;

    const float* ap = x_pts + ((size_t)(n * P1_PTS + i0 + l15)) * 3;
    v2f afrag;
    afrag.x = half ? ap[2] : ap[0];
    afrag.y = half ? 0.0f  : ap[1];

    const int col   = j0 + l15;
    const float* bp = y_pts + ((size_t)(n * P2_PTS + col)) * 3;
    const float b0 = bp[0], b1 = bp[1], b2 = bp[2];
    v2f bfrag;
    bfrag.x = half ? b2   : b0;
    bfrag.y = half ? 0.0f : b1;
    const float yn = b0 * b0 + b1 * b1 + b2 * b2;
    const float vc = v[n * P2_PTS + col];

    const v8f dotc = wmma_dot16x16(afrag, bfrag);

    float local = 0.0f;
#pragma unroll
    for (int g = 0; g < 8; ++g) {
        const int row  = i0 + g + half * 8;
        const float* p = x_pts + ((size_t)(n * P1_PTS + row)) * 3;
        const float xn  = p[0] * p[0] + p[1] * p[1] + p[2] * p[2];
        const float Cij = xn + yn - 2.0f * dotc[g];
        const float Mv  = (u[n * P1_PTS + row] + vc - Cij) * INV_EPS_F;
        const float piv = __expf(Mv);
        const size_t off = ((size_t)n * P1_PTS + row) * P2_PTS + col;
        pi_out[off] = piv;
        c_out[off]  = Cij;
        local += piv * Cij;
    }
#pragma unroll
    for (int off = 16; off >= 1; off >>= 1)
        local += __shfl_xor(local, off, 32);
    if (lane == 0) atomicAdd(cost, local);
}

// ---------------------------------------------------------------------------
extern "C" void kernel_launch(void* const* d_in, const int* in_sizes, int n_in,
                              void* d_out, int out_size, void* d_ws,
                              size_t ws_size, hipStream_t stream) {
    (void)in_sizes; (void)n_in; (void)out_size; (void)ws_size;

    const float* x  = (const float*)d_in[0];  // [N, P1, 3]
    const float* y  = (const float*)d_in[1];  // [N, P2, 3]
    const float* wx = (const float*)d_in[2];  // [N, P1]
    const float* wy = (const float*)d_in[3];  // [N, P2]

    float* out  = (float*)d_out;
    float* cost = out;                                     // scalar
    float* pi   = out + 1;                                 // [N, P1, P2]
    float* cmat = pi + (size_t)N_BATCH * P1_PTS * P2_PTS;  // [N, P1, P2]

    float* u   = (float*)d_ws;               // [N, P1]
    float* v   = u + N_BATCH * P1_PTS;       // [N, P2]
    float* err = v + N_BATCH * P2_PTS;       // [MAX_ITER]
    int*   done = (int*)(err + MAX_ITER);    // [1]

    {
        const int tot = N_BATCH * P1_PTS;
        sinkhorn_init<<<(tot + 255) / 256, 256, 0, stream>>>(u, v, err, done,
                                                             cost);
    }

    const dim3 blk(WAVES_PER_BLOCK * 32);
    const dim3 grid_row(P1_PTS / (16 * WAVES_PER_BLOCK), N_BATCH);
    const dim3 grid_col(P2_PTS / (16 * WAVES_PER_BLOCK), N_BATCH);
    const float thresh_total = THRESH_F * (float)N_BATCH;  // mean over N

    for (int it = 0; it < MAX_ITER; ++it) {
        // u update (accumulates convergence error)
        sinkhorn_lse_pass<<<grid_row, blk, 0, stream>>>(
            x, y, wx, v, u, err + it, done, P1_PTS, P2_PTS, 1);
        // v update (uses the fresh u; C symmetric under x<->y swap)
        sinkhorn_lse_pass<<<grid_col, blk, 0, stream>>>(
            y, x, wy, u, v, err + it, done, P2_PTS, P1_PTS, 0);
        sinkhorn_flag<<<1, 32, 0, stream>>>(err, it, done, thresh_total);
    }

    const dim3 grid_fin(P2_PTS / 16, P1_PTS / 16, N_BATCH);
    sinkhorn_final<<<grid_fin, dim3(32), 0, stream>>>(x, y, u, v, pi, cmat,
                                                      cost);
}